// AgriGraphGCN_87471303950804
// MI455X (gfx1250) — compile-verified
//
#include <hip/hip_runtime.h>
#include <math.h>

// ---------------------------------------------------------------------------
// AgriGraphGCN: 3-layer GCN (GCNConv -> BN -> ReLU) x2 -> GCNConv -> sigmoid
// N=100000 nodes, E=640000 edges, IN=6, H=128, OUT=1, all f32.
//
// MI455X reasoning: scatter/gather over the 51 MB feature matrix dominates
// (L2-resident on the 192 MB L2). Compute is trivial (~3.3 GFLOP), so we keep
// full f32 precision and use V_WMMA_F32_16X16X4_F32 for the 128x128 GEMM.
// ---------------------------------------------------------------------------

typedef float v2f __attribute__((ext_vector_type(2)));
typedef float v8f __attribute__((ext_vector_type(8)));

#define HDIM 128
#define BN_EPS 1e-5f

// ------------------------------- utility -----------------------------------
__global__ void k_fill(float* p, float v, int n) {
  int i = blockIdx.x * blockDim.x + threadIdx.x;
  if (i < n) p[i] = v;
}

// deg[dst] += 1 per edge (deg buffer pre-filled with 1.0 for the self-loop)
__global__ void k_deg_edges(const int* __restrict__ dst, float* __restrict__ deg, int E) {
  int e = blockIdx.x * blockDim.x + threadIdx.x;
  if (e < E) atomicAdd(&deg[dst[e]], 1.0f);
}

// deg -> dis = rsqrt(deg) in place
__global__ void k_rsqrt_inplace(float* p, int n) {
  int i = blockIdx.x * blockDim.x + threadIdx.x;
  if (i < n) {
    float d = p[i];
    p[i] = (d > 0.0f) ? rsqrtf(d) : 0.0f;
  }
}

// ------------------------- layer 1: x(N,6) @ W1(6,128) ----------------------
__global__ void k_lin1(const float* __restrict__ x, const float* __restrict__ W1,
                       float* __restrict__ out, int N) {
  int t = blockIdx.x * blockDim.x + threadIdx.x;
  if (t >= N * HDIM) return;
  int i = t >> 7;        // node
  int f = t & (HDIM - 1); // feature
  const float* xr = x + (long)i * 6;
  float acc = 0.0f;
#pragma unroll
  for (int k = 0; k < 6; ++k) acc = fmaf(xr[k], W1[k * HDIM + f], acc);
  out[t] = acc;
}

// ------------------------- aggregation (128-wide) ---------------------------
// agg[i,:] = m[i,:] * dis[i]^2   (self-loop term, also zero-initializes agg)
__global__ void k_selfloop_init(const float* __restrict__ m, const float* __restrict__ dis,
                                float* __restrict__ agg, int N) {
  int t = blockIdx.x * blockDim.x + threadIdx.x;
  if (t >= N * HDIM) return;
  float d = dis[t >> 7];
  agg[t] = m[t] * d * d;
}

// one wave per edge: agg[dst,:] += m[src,:] * dis[src]*dis[dst]
__global__ void k_agg_edges(const int* __restrict__ src, const int* __restrict__ dst,
                            const float* __restrict__ dis, const float* __restrict__ m,
                            float* __restrict__ agg, int E) {
  int wave = (int)((blockIdx.x * blockDim.x + threadIdx.x) >> 5);
  int lane = threadIdx.x & 31;
  if (wave >= E) return;
  int s = src[wave], d = dst[wave];
  float nrm = dis[s] * dis[d];
  const float* ms = m + (long)s * HDIM;
  float* ad = agg + (long)d * HDIM;
#pragma unroll
  for (int j = 0; j < 4; ++j) {
    int f = lane + j * 32;
    atomicAdd(&ad[f], ms[f] * nrm);
  }
}

// ------------------------------- batch norm ---------------------------------
// per-feature partial sums: blockDim.x == 128, thread f owns feature f
__global__ void k_stats(const float* __restrict__ h, float* __restrict__ gsum,
                        float* __restrict__ gsumsq, int N) {
  int f = threadIdx.x;
  float s = 0.0f, ss = 0.0f;
  for (int i = blockIdx.x; i < N; i += gridDim.x) {
    float v = h[(long)i * HDIM + f];
    s += v;
    ss += v * v;
  }
  atomicAdd(&gsum[f], s);
  atomicAdd(&gsumsq[f], ss);
}

// fold BN into scale/shift: y = v*scale + shift ; (conv bias cancels in BN)
__global__ void k_stats_finalize(const float* __restrict__ gsum, const float* __restrict__ gsumsq,
                                 const float* __restrict__ g, const float* __restrict__ be,
                                 float* __restrict__ scale, float* __restrict__ shift, int N) {
  int f = threadIdx.x;
  if (f >= HDIM) return;
  float inv = 1.0f / (float)N;
  float mu = gsum[f] * inv;
  float var = gsumsq[f] * inv - mu * mu;  // biased variance, as in reference
  float rstd = rsqrtf(var + BN_EPS);
  float sc = g[f] * rstd;
  scale[f] = sc;
  shift[f] = be[f] - mu * sc;
}

// in-place BN + ReLU
__global__ void k_bnrelu(float* __restrict__ hbuf, const float* __restrict__ scale,
                         const float* __restrict__ shift, int N) {
  int t = blockIdx.x * blockDim.x + threadIdx.x;
  if (t >= N * HDIM) return;
  int f = t & (HDIM - 1);
  float v = fmaf(hbuf[t], scale[f], shift[f]);
  hbuf[t] = v > 0.0f ? v : 0.0f;
}

// ------------------- layer 2 GEMM: C(N,128) = A(N,128) @ B(128,128) ---------
// One wave per 16x16 output tile; 32x V_WMMA_F32_16X16X4_F32 along K.
// Fragment layouts per CDNA5 ISA 7.12.2 (32-bit A 16x4, 32-bit B 4x16, C 16x16).
__global__ void __launch_bounds__(256)
k_gemm_wmma(const float* __restrict__ A, const float* __restrict__ B,
            float* __restrict__ C, int N) {
  int wave = threadIdx.x >> 5;
  int lane = threadIdx.x & 31;
  int tile = blockIdx.x * (blockDim.x >> 5) + wave;
  int nRowTiles = N >> 4;                 // N is a multiple of 16 (100000/16 = 6250)
  int rt = tile >> 3;                     // 8 column tiles (128/16)
  int ct = tile & 7;
  if (rt >= nRowTiles) return;            // uniform across the wave -> EXEC stays all-1s
  int rb = rt << 4;
  int cb = ct << 4;

  int m    = lane & 15;                   // row within tile (A) / col within tile (B,C)
  int koff = (lane >> 4) << 1;            // lanes 16-31 hold K=+2,+3

  v8f c = {};                             // 16x16 f32 accumulator (8 VGPRs)
  const float* arow = A + (long)(rb + m) * HDIM + koff;
  const float* bcol = B + cb + m;         // column index == lane&15

#pragma unroll
  for (int k0 = 0; k0 < HDIM; k0 += 4) {
    v2f a = *(const v2f*)(arow + k0);     // A[m][k0+koff], A[m][k0+koff+1] (8B aligned)
    v2f b;
    b.x = bcol[(k0 + koff) * HDIM];       // B[k0+koff][n]
    b.y = bcol[(k0 + koff + 1) * HDIM];   // B[k0+koff+1][n]
    c = __builtin_amdgcn_wmma_f32_16x16x4_f32(
        /*neg_a=*/false, a, /*neg_b=*/false, b,
        /*c_mod=*/(short)0, c, /*reuse_a=*/false, /*reuse_b=*/false);
  }

  // C layout: VGPR r -> row rb + r + 8*(lane>>4), col cb + (lane&15)
  float* crow = C + (long)(rb + ((lane >> 4) << 3)) * HDIM + cb + m;
#pragma unroll
  for (int r = 0; r < 8; ++r) crow[r * HDIM] = c[r];
}

// ---------------------- layer 3: h(N,128) @ W3(128,1) -----------------------
__global__ void k_dot_w3(const float* __restrict__ h, const float* __restrict__ W3,
                         float* __restrict__ m3, int N) {
  int wave = (int)((blockIdx.x * blockDim.x + threadIdx.x) >> 5);
  int lane = threadIdx.x & 31;
  if (wave >= N) return;
  const float* hr = h + (long)wave * HDIM;
  float s = 0.0f;
#pragma unroll
  for (int j = 0; j < 4; ++j) {
    int f = lane + 32 * j;
    s = fmaf(hr[f], W3[f], s);
  }
#pragma unroll
  for (int off = 16; off; off >>= 1) s += __shfl_xor(s, off, 32);
  if (lane == 0) m3[wave] = s;
}

// scalar (1-wide) aggregation
__global__ void k_agg3_init(const float* __restrict__ m3, const float* __restrict__ dis,
                            float* __restrict__ a3, int N) {
  int i = blockIdx.x * blockDim.x + threadIdx.x;
  if (i < N) {
    float d = dis[i];
    a3[i] = m3[i] * d * d;
  }
}

__global__ void k_agg3_edges(const int* __restrict__ src, const int* __restrict__ dst,
                             const float* __restrict__ dis, const float* __restrict__ m3,
                             float* __restrict__ a3, int E) {
  int e = blockIdx.x * blockDim.x + threadIdx.x;
  if (e < E) {
    int s = src[e], d = dst[e];
    atomicAdd(&a3[d], m3[s] * dis[s] * dis[d]);
  }
}

__global__ void k_out_sigmoid(const float* __restrict__ a3, const float* __restrict__ b3,
                              float* __restrict__ out, int N) {
  int i = blockIdx.x * blockDim.x + threadIdx.x;
  if (i < N) {
    float z = a3[i] + b3[0];
    out[i] = 1.0f / (1.0f + __expf(-z));
  }
}

// ---------------------------------------------------------------------------
static inline size_t align256(size_t x) { return (x + 255u) & ~(size_t)255u; }

extern "C" void kernel_launch(void* const* d_in, const int* in_sizes, int n_in,
                              void* d_out, int out_size, void* d_ws, size_t ws_size,
                              hipStream_t stream) {
  // Inputs in setup_inputs() order:
  const float* x   = (const float*)d_in[0];   // (N,6)
  const int*   ei  = (const int*)d_in[1];     // (2,E) edge_index
  const float* W1  = (const float*)d_in[2];   // (6,128)
  // d_in[3] = b1 (cancels in BN)
  const float* W2  = (const float*)d_in[4];   // (128,128)
  // d_in[5] = b2 (cancels in BN)
  const float* W3  = (const float*)d_in[6];   // (128,1)
  const float* b3  = (const float*)d_in[7];   // (1,)
  const float* g1  = (const float*)d_in[8];
  const float* be1 = (const float*)d_in[9];
  const float* g2  = (const float*)d_in[10];
  const float* be2 = (const float*)d_in[11];

  const int N = in_sizes[0] / 6;
  const int E = in_sizes[1] / 2;
  const int* src = ei;        // edge_index[0,:]
  const int* dst = ei + E;    // edge_index[1,:]

  // Workspace carve-up (two ping-pong N x 128 buffers keep working set in L2)
  char* ws = (char*)d_ws;
  size_t nbH = align256((size_t)N * HDIM * sizeof(float));
  size_t nb1 = align256((size_t)N * sizeof(float));
  float* B0    = (float*)(ws);                 // N x 128
  float* B1    = (float*)(ws + nbH);           // N x 128
  float* dis   = (float*)(ws + 2 * nbH);       // N   (deg -> dis)
  float* m3    = (float*)(ws + 2 * nbH + nb1); // N
  float* a3    = (float*)(ws + 2 * nbH + 2 * nb1); // N
  float* stats = (float*)(ws + 2 * nbH + 3 * nb1); // 512 floats:
  float* gsum  = stats;        // [0,128)
  float* gsq   = stats + 128;  // [128,256)
  float* scl   = stats + 256;  // [256,384)
  float* shf   = stats + 384;  // [384,512)

  const int TPB = 256;
  int gN   = (N + TPB - 1) / TPB;
  int gNH  = (N * HDIM + TPB - 1) / TPB;
  int gE   = (E + TPB - 1) / TPB;
  int gEw  = (E + (TPB / 32) - 1) / (TPB / 32);   // one wave per edge
  int gNw  = (N + (TPB / 32) - 1) / (TPB / 32);   // one wave per node
  int nTiles = (N >> 4) * (HDIM >> 4);
  int gT   = (nTiles + (TPB / 32) - 1) / (TPB / 32);

  // --- degree / normalization -------------------------------------------------
  k_fill<<<gN, TPB, 0, stream>>>(dis, 1.0f, N);                       // self-loop
  k_deg_edges<<<gE, TPB, 0, stream>>>(dst, dis, E);
  k_rsqrt_inplace<<<gN, TPB, 0, stream>>>(dis, N);

  // --- layer 1: linear -> aggregate -> BN -> ReLU -----------------------------
  k_lin1<<<gNH, TPB, 0, stream>>>(x, W1, B0, N);
  k_selfloop_init<<<gNH, TPB, 0, stream>>>(B0, dis, B1, N);
  k_agg_edges<<<gEw, TPB, 0, stream>>>(src, dst, dis, B0, B1, E);
  k_fill<<<1, 256, 0, stream>>>(gsum, 0.0f, 256);                     // zero sum+sumsq
  k_stats<<<1024, HDIM, 0, stream>>>(B1, gsum, gsq, N);
  k_stats_finalize<<<1, HDIM, 0, stream>>>(gsum, gsq, g1, be1, scl, shf, N);
  k_bnrelu<<<gNH, TPB, 0, stream>>>(B1, scl, shf, N);

  // --- layer 2: WMMA GEMM -> aggregate -> BN -> ReLU --------------------------
  k_gemm_wmma<<<gT, TPB, 0, stream>>>(B1, W2, B0, N);
  k_selfloop_init<<<gNH, TPB, 0, stream>>>(B0, dis, B1, N);
  k_agg_edges<<<gEw, TPB, 0, stream>>>(src, dst, dis, B0, B1, E);
  k_fill<<<1, 256, 0, stream>>>(gsum, 0.0f, 256);
  k_stats<<<1024, HDIM, 0, stream>>>(B1, gsum, gsq, N);
  k_stats_finalize<<<1, HDIM, 0, stream>>>(gsum, gsq, g2, be2, scl, shf, N);
  k_bnrelu<<<gNH, TPB, 0, stream>>>(B1, scl, shf, N);

  // --- layer 3: dot with W3 -> aggregate -> +b3 -> sigmoid --------------------
  k_dot_w3<<<gNw, TPB, 0, stream>>>(B1, W3, m3, N);
  k_agg3_init<<<gN, TPB, 0, stream>>>(m3, dis, a3, N);
  k_agg3_edges<<<gE, TPB, 0, stream>>>(src, dst, dis, m3, a3, E);
  k_out_sigmoid<<<gN, TPB, 0, stream>>>(a3, b3, (float*)d_out, N);
}